// CovBlock_4887672783552
// MI455X (gfx1250) — compile-verified
//
#include <hip/hip_runtime.h>

typedef __attribute__((ext_vector_type(2))) float v2f;
typedef __attribute__((ext_vector_type(8))) float v8f;

#define B_ 4
#define N_ 8192
#define D_ 2048
#define H_ 512
#define F_ 64
#define EPSV 1e-6f
#define NCHUNK 64
#define ROWS_PER_CHUNK (N_ / NCHUNK) /* 128 */

// ---------------------------------------------------------------------------
// Pass 1: per-row mean over the feature axis. One 256-thread block per row.
// Coalesced: thread t reads x[row*D + t + 256*i].
// ---------------------------------------------------------------------------
__global__ void rowmean_kernel(const float* __restrict__ x,
                               float* __restrict__ means) {
    __shared__ float red[256];
    const int row = blockIdx.x; // 0 .. B*N-1
    const float* xr = x + (size_t)row * D_;
    float s = 0.f;
    for (int d = threadIdx.x; d < D_; d += 256) s += xr[d];
    red[threadIdx.x] = s;
    __syncthreads();
    for (int off = 128; off > 0; off >>= 1) {
        if ((int)threadIdx.x < off) red[threadIdx.x] += red[threadIdx.x + off];
        __syncthreads();
    }
    if (threadIdx.x == 0) means[row] = red[0] * (1.0f / (float)D_);
}

// ---------------------------------------------------------------------------
// Pass 2: partial sum of (x - mean)^2 over an N-chunk, per column.
// Thread -> column (coalesced b32 loads). Deterministic (no FP atomics):
// partial[chunk][b][d].
// ---------------------------------------------------------------------------
__global__ void colssq_kernel(const float* __restrict__ x,
                              const float* __restrict__ means,
                              float* __restrict__ partial) {
    const int dt = blockIdx.x;    // 0 .. D/256-1
    const int chunk = blockIdx.y; // 0 .. NCHUNK-1
    const int b = blockIdx.z;     // 0 .. B-1
    const int d = dt * 256 + threadIdx.x;
    const int n0 = chunk * ROWS_PER_CHUNK;
    const float* xb = x + ((size_t)b * N_ + n0) * D_ + d;
    const float* mb = means + (size_t)b * N_ + n0;
    float acc = 0.f;
#pragma unroll 4
    for (int n = 0; n < ROWS_PER_CHUNK; ++n) {
        float v = xb[(size_t)n * D_] - mb[n];
        acc = fmaf(v, v, acc);
    }
    partial[((size_t)chunk * B_ + b) * D_ + d] = acc;
}

// ---------------------------------------------------------------------------
// Reduce partials -> cov_diag = ss/(ss+eps); emit padded A1[16][D]
// (rows >= B are zero so WMMA can run a full 16-row M tile).
// ---------------------------------------------------------------------------
__global__ void cov_kernel(const float* __restrict__ partial,
                           float* __restrict__ A1) {
    const int idx = blockIdx.x * 256 + threadIdx.x; // 0 .. 16*D-1
    const int r = idx / D_;
    const int d = idx - r * D_;
    float v = 0.f;
    if (r < B_) {
        float ss = 0.f;
        for (int c = 0; c < NCHUNK; ++c)
            ss += partial[((size_t)c * B_ + r) * D_ + d];
        v = ss / (ss + EPSV);
    }
    A1[idx] = v;
}

// ---------------------------------------------------------------------------
// FP32 WMMA GEMM: Out[m,n] = act(A[16,K] @ W[K,N] + bias[n]), act = leakyReLU
// (slope==1.0 -> identity). One wave per 16x16 output tile, K stepped by 4
// with V_WMMA_F32_16X16X4_F32. K and N are compile-time constants so the
// W row stride folds into immediate global_load offsets (one pointer bump
// per unrolled group instead of per-load 64-bit address math).
//
// ISA 7.12.2 fp32 operand layouts (wave32):
//   A 16x4 : VGPR0 = {K0 | lanes 0-15, K2 | lanes 16-31}, VGPR1 = {K1, K3};
//            M = lane & 15.
//   B 4x16 : VGPR0 = {K0 | lanes 0-15, K2 | lanes 16-31}, VGPR1 = {K1, K3};
//            N = lane & 15.
//   C/D    : VGPR r -> row (r + 8*(lane>=16)), col = lane & 15.
// ---------------------------------------------------------------------------
template <int K, int N>
__global__ void wmma_gemm_kernel(const float* __restrict__ A,    // [16,K]
                                 const float* __restrict__ W,    // [K,N]
                                 const float* __restrict__ bias, // [N]
                                 float* __restrict__ Out,        // [Mstore,ldOut]
                                 int Mstore, int ldOut, float slope) {
    const int lane = threadIdx.x & 31;
    const int wave = threadIdx.x >> 5;
    const int tile = blockIdx.x * (blockDim.x >> 5) + wave;
    constexpr int ntiles = N >> 4;
    if (tile >= ntiles) return; // uniform per wave: EXEC stays all-ones

    const int n0 = tile << 4;
    const int hi = lane >> 4;   // 0 | 1
    const int l16 = lane & 15;  // row (A) / col (B,C,D) within tile
    const int koff = hi << 1;   // 0 | 2

    const float* arow = A + (size_t)l16 * K + koff;
    const float* wcol = W + (size_t)koff * N + n0 + l16;

    v8f acc = {0.f, 0.f, 0.f, 0.f, 0.f, 0.f, 0.f, 0.f};
#pragma unroll 4
    for (int kk = 0; kk < K; kk += 4) {
        // A pair: contiguous in K -> single b64 load, immediate offset.
        v2f a = {arow[kk], arow[kk + 1]};
        // B pair: rows kk+koff, kk+koff+1; constant row stride N -> imm offs.
        v2f b = {wcol[(size_t)kk * N], wcol[(size_t)(kk + 1) * N]};
        acc = __builtin_amdgcn_wmma_f32_16x16x4_f32(
            /*neg_a=*/false, a, /*neg_b=*/false, b,
            /*c_mod=*/(short)0, acc, /*reuse_a=*/false, /*reuse_b=*/false);
    }

    const int col = n0 + l16;
    const float bv = bias[col];
    const int mbase = hi << 3; // 0 | 8
#pragma unroll
    for (int r = 0; r < 8; ++r) {
        const int m = mbase + r;
        float v = acc[r] + bv;
        v = (v >= 0.f) ? v : v * slope;
        if (m < Mstore) Out[(size_t)m * ldOut + col] = v;
    }
}

// ---------------------------------------------------------------------------
extern "C" void kernel_launch(void* const* d_in, const int* in_sizes, int n_in,
                              void* d_out, int out_size, void* d_ws,
                              size_t ws_size, hipStream_t stream) {
    (void)in_sizes; (void)n_in; (void)out_size; (void)ws_size;
    const float* x  = (const float*)d_in[0];
    const float* W1 = (const float*)d_in[1];
    const float* b1 = (const float*)d_in[2];
    const float* W2 = (const float*)d_in[3];
    const float* b2 = (const float*)d_in[4];
    const float* W3 = (const float*)d_in[5];
    const float* b3 = (const float*)d_in[6];
    float* out = (float*)d_out;

    float* ws      = (float*)d_ws;
    float* means   = ws;                                   // B*N      = 32768
    float* partial = means + (size_t)B_ * N_;              // 64*B*D   = 524288
    float* A1      = partial + (size_t)NCHUNK * B_ * D_;   // 16*D     = 32768
    float* h1      = A1 + 16 * D_;                         // 16*D     = 32768
    float* h2      = h1 + 16 * D_;                         // 16*H     = 8192

    // Pass 1: row means over D.
    rowmean_kernel<<<B_ * N_, 256, 0, stream>>>(x, means);

    // Pass 2: per-column sum of squares, chunked over N (deterministic).
    dim3 g2(D_ / 256, NCHUNK, B_);
    colssq_kernel<<<g2, 256, 0, stream>>>(x, means, partial);

    // cov_diag -> padded [16, D] activation matrix.
    cov_kernel<<<(16 * D_) / 256, 256, 0, stream>>>(partial, A1);

    // 3-layer MLP, fp32 WMMA. 8 waves/block (4 in layer 3), 1 tile/wave.
    wmma_gemm_kernel<D_, D_><<<(D_ / 16 + 7) / 8, 256, 0, stream>>>(
        A1, W1, b1, h1, 16, D_, 0.01f);
    wmma_gemm_kernel<D_, H_><<<(H_ / 16 + 7) / 8, 256, 0, stream>>>(
        h1, W2, b2, h2, 16, H_, 0.01f);
    wmma_gemm_kernel<H_, F_><<<1, 128, 0, stream>>>(
        h2, W3, b3, out, B_, F_, 1.0f);
}